// VariableSpatialTransformer_30992484008613
// MI455X (gfx1250) — compile-verified
//
#include <hip/hip_runtime.h>

#define AS1 __attribute__((address_space(1)))
#define AS3 __attribute__((address_space(3)))

typedef int v4i __attribute__((ext_vector_type(4)));

namespace {
constexpr int Bc   = 4;
constexpr int Cc   = 2;
constexpr int Hc   = 1024;
constexpr int Wc   = 1024;
constexpr int HW   = Hc * Wc;
constexpr int ROWS = 4;      // rows per block (double-buffered async pipeline)
constexpr int TPB  = 256;    // 8 wave32 waves
}

// ---- CDNA5 async-to-LDS copy of 16 bytes (per-lane), tracked by ASYNCcnt ----
__device__ __forceinline__ void async_cp16(const float* __restrict__ g, float* l) {
#if defined(__AMDGCN__)
#if __has_builtin(__builtin_amdgcn_global_load_async_to_lds_b128)
  __builtin_amdgcn_global_load_async_to_lds_b128(
      (AS1 v4i*)g, (AS3 v4i*)l, /*offset=*/0, /*cpol=*/0);
#else
  unsigned lds_addr = (unsigned)(unsigned long long)l;  // low 32 bits = LDS byte addr
  asm volatile("global_load_async_to_lds_b128 %0, %1, off"
               :: "v"(lds_addr), "v"(g)
               : "memory");
#endif
#else
  (void)g; (void)l;
#endif
}

#if defined(__AMDGCN__)
#if __has_builtin(__builtin_amdgcn_s_wait_asynccnt)
#define WAIT_ASYNC(n) __builtin_amdgcn_s_wait_asynccnt(n)
#else
#define WAIT_ASYNC(n) asm volatile("s_wait_asynccnt %0" :: "i"(n) : "memory")
#endif
#else
#define WAIT_ASYNC(n) do { } while (0)
#endif

__global__ __launch_bounds__(TPB)
void vst_bilinear_kernel(const float* __restrict__ src,
                         const float* __restrict__ flow,
                         float* __restrict__ out) {
  // Double-buffered per-row flow staging: 2 bufs x 2 planes x 1024 f32 = 16 KB
  __shared__ alignas(16) float ldsY[2][Wc];
  __shared__ alignas(16) float ldsX[2][Wc];

  const int g0 = blockIdx.x * ROWS;       // first global row handled by block
  const int b  = g0 >> 10;                // / H (H = 1024)
  const int y0 = g0 & (Hc - 1);
  const int xi = (int)threadIdx.x * 4;    // 4 pixels per thread -> full 1024 row

  const float* __restrict__ flowY = flow + ((size_t)b * 2 + 0) * HW;
  const float* __restrict__ flowX = flow + ((size_t)b * 2 + 1) * HW;
  const float* __restrict__ srcb  = src + (size_t)b * Cc * HW;   // channel-0 plane
  float*       __restrict__ outb  = out + (size_t)b * Cc * HW;

  // Prologue: stage flow for row y0 (2 async b128 per lane -> ASYNCcnt += 2)
  async_cp16(flowY + (size_t)y0 * Wc + xi, &ldsY[0][xi]);
  async_cp16(flowX + (size_t)y0 * Wc + xi, &ldsX[0][xi]);

#pragma unroll
  for (int r = 0; r < ROWS; ++r) {
    const int y   = y0 + r;
    const int buf = r & 1;

    if (r + 1 < ROWS) {
      // Stage next row into the other buffer, then wait only for the 2 older
      // copies (ASYNCcnt decrements in issue order for async loads).
      async_cp16(flowY + (size_t)(y + 1) * Wc + xi, &ldsY[buf ^ 1][xi]);
      async_cp16(flowX + (size_t)(y + 1) * Wc + xi, &ldsX[buf ^ 1][xi]);
      WAIT_ASYNC(2);
    } else {
      WAIT_ASYNC(0);
    }

    // Each lane reads back exactly the 16B it staged: no cross-lane, no barrier.
    const float4 fy4 = *reinterpret_cast<const float4*>(&ldsY[buf][xi]);
    const float4 fx4 = *reinterpret_cast<const float4*>(&ldsX[buf][xi]);
    const float fys[4] = {fy4.x, fy4.y, fy4.z, fy4.w};
    const float fxs[4] = {fx4.x, fx4.y, fx4.z, fx4.w};

    float acc0[4], acc1[4];
#pragma unroll
    for (int j = 0; j < 4; ++j) {
      const float sy = (float)y + fys[j];
      const float sx = (float)(xi + j) + fxs[j];
      const float yf = floorf(sy);
      const float xf = floorf(sx);
      const float wy = sy - yf;
      const float wx = sx - xf;
      const int iy0 = (int)yf, ix0 = (int)xf;
      const int iy1 = iy0 + 1, ix1 = ix0 + 1;

      // zeros padding: mask invalid corners (exact reference semantics)
      const float my0 = ((unsigned)iy0 < (unsigned)Hc) ? 1.0f : 0.0f;
      const float my1 = ((unsigned)iy1 < (unsigned)Hc) ? 1.0f : 0.0f;
      const float mx0 = ((unsigned)ix0 < (unsigned)Wc) ? 1.0f : 0.0f;
      const float mx1 = ((unsigned)ix1 < (unsigned)Wc) ? 1.0f : 0.0f;

      const int cy0 = min(Hc - 1, max(0, iy0));
      const int cy1 = min(Hc - 1, max(0, iy1));
      const int cx0 = min(Wc - 1, max(0, ix0));
      const int cx1 = min(Wc - 1, max(0, ix1));

      const int i00 = cy0 * Wc + cx0;
      const int i01 = cy0 * Wc + cx1;
      const int i10 = cy1 * Wc + cx0;
      const int i11 = cy1 * Wc + cx1;

      const float wyc = 1.0f - wy;
      const float wxc = 1.0f - wx;
      const float w00 = wyc * wxc * (my0 * mx0);
      const float w01 = wyc * wx  * (my0 * mx1);
      const float w10 = wy  * wxc * (my1 * mx0);
      const float w11 = wy  * wx  * (my1 * mx1);

      // Both channels share indices & weights; src planes are HW apart.
      const float* __restrict__ p0 = srcb;
      const float* __restrict__ p1 = srcb + HW;
      acc0[j] = fmaf(p0[i00], w00,
                fmaf(p0[i01], w01,
                fmaf(p0[i10], w10, p0[i11] * w11)));
      acc1[j] = fmaf(p1[i00], w00,
                fmaf(p1[i01], w01,
                fmaf(p1[i10], w10, p1[i11] * w11)));
    }

    const float4 o0 = {acc0[0], acc0[1], acc0[2], acc0[3]};
    const float4 o1 = {acc1[0], acc1[1], acc1[2], acc1[3]};
    *reinterpret_cast<float4*>(outb + (size_t)y * Wc + xi)              = o0;
    *reinterpret_cast<float4*>(outb + (size_t)HW + (size_t)y * Wc + xi) = o1;
  }
}

extern "C" void kernel_launch(void* const* d_in, const int* in_sizes, int n_in,
                              void* d_out, int out_size, void* d_ws, size_t ws_size,
                              hipStream_t stream) {
  (void)in_sizes; (void)n_in; (void)out_size; (void)d_ws; (void)ws_size;
  const float* src  = (const float*)d_in[0];   // [4,2,1024,1024] f32
  const float* flow = (const float*)d_in[1];   // [4,2,1024,1024] f32
  float*       out  = (float*)d_out;           // [4,2,1024,1024] f32

  dim3 grid(Bc * Hc / ROWS);   // 1024 blocks, each: 4 rows x 1024 px x 2 ch
  dim3 block(TPB);             // 256 threads = 8 wave32
  vst_bilinear_kernel<<<grid, block, 0, stream>>>(src, flow, out);
}